// SampledSoftmax_73057393705216
// MI455X (gfx1250) — compile-verified
//
#include <hip/hip_runtime.h>
#include <hip/hip_bf16.h>
#include <math.h>

typedef __attribute__((ext_vector_type(2))) float v2f;
typedef __attribute__((ext_vector_type(4))) float v4f;
typedef __attribute__((ext_vector_type(8))) float v8f;

#define NROWS     8192
#define HIDDIM    1024
#define NSAMP     8192
#define OUTSTRIDE (NSAMP + 1)
#define NEG_INF_V (-1e37f)

// GEMM tiling
#define BM   128
#define BN   128
#define KC   32
#define KPAD (KC + 2)   // +2 floats: keeps (row*KPAD + evenK)*4 8B-aligned, no LDS bank conflicts

// ---------------------------------------------------------------------------
// Main kernel: logits[n, 1+s] = dot(output[n], weight[sample_ids[s]])
//                              + bias[sid] - log(sample_freq[s]),
//              masked to -1e37 where targets[n] == sample_ids[s].
// Uses V_WMMA_F32_16X16X4_F32 (full fp32 precision on the matrix pipe).
// ---------------------------------------------------------------------------
__global__ __launch_bounds__(256) void sampled_gemm_kernel(
    const float* __restrict__ outp,
    const int*   __restrict__ targets,
    const int*   __restrict__ sample_ids,
    const float* __restrict__ sample_freq,
    const float* __restrict__ weight,
    const float* __restrict__ bias,
    float*       __restrict__ logits)
{
    __shared__ float A_s[BM * KPAD];
    __shared__ float B_s[BN * KPAD];
    __shared__ int   sid_s[BN];
    __shared__ int   tgt_s[BM];

    const int tid     = threadIdx.x;
    const int lane    = tid & 31;
    const int wave    = tid >> 5;
    const int rowBase = blockIdx.y * BM;
    const int colBase = blockIdx.x * BN;

    // Stage gathered ids and the target tile once.
    if (tid < BN)                 sid_s[tid]        = sample_ids[colBase + tid];
    if (tid >= 128 && tid < 256)  tgt_s[tid - 128]  = targets[rowBase + tid - 128];
    __syncthreads();

    // Global->LDS assignment: each thread owns 4 float4 chunks.
    const int rsub = tid >> 3;       // 0..31 (tile row, +32*i)
    const int cg   = (tid & 7) * 4;  // k-offset within the K-slice
    int grow[4];
#pragma unroll
    for (int i = 0; i < 4; ++i) grow[i] = sid_s[rsub + 32 * i];

    // Wave tile: 32 rows x 64 cols  (2 x 4 WMMA 16x16 tiles)
    const int waveM = (wave & 3) * 32;    // 0,32,64,96
    const int waveN = (wave >> 2) * 64;   // 0,64
    const int ml    = lane & 15;
    const int khalf = (lane < 16) ? 0 : 2;   // fp32 A/B fragment K striping

    v8f acc[2][4];
#pragma unroll
    for (int i = 0; i < 2; ++i)
#pragma unroll
        for (int j = 0; j < 4; ++j)
            acc[i][j] = 0.0f;

    for (int kb = 0; kb < HIDDIM; kb += KC) {
        // ---- stage A (activations) ----
#pragma unroll
        for (int i = 0; i < 4; ++i) {
            const int r = rsub + 32 * i;
            const v4f av = *(const v4f*)(outp + (size_t)(rowBase + r) * HIDDIM + kb + cg);
            v2f lo = {av.x, av.y};
            v2f hi = {av.z, av.w};
            *(v2f*)&A_s[r * KPAD + cg]     = lo;
            *(v2f*)&A_s[r * KPAD + cg + 2] = hi;
        }
        // ---- stage B (gathered weight rows) ----
#pragma unroll
        for (int i = 0; i < 4; ++i) {
            const int r = rsub + 32 * i;
            const v4f wv = *(const v4f*)(weight + (size_t)grow[i] * HIDDIM + kb + cg);
            v2f lo = {wv.x, wv.y};
            v2f hi = {wv.z, wv.w};
            *(v2f*)&B_s[r * KPAD + cg]     = lo;
            *(v2f*)&B_s[r * KPAD + cg + 2] = hi;
        }
        // Prefetch next K-slice while this one is consumed.
        if (kb + KC < HIDDIM) {
            __builtin_prefetch(outp + (size_t)(rowBase + rsub) * HIDDIM + kb + KC + cg, 0, 1);
            __builtin_prefetch(weight + (size_t)grow[0] * HIDDIM + kb + KC + cg, 0, 1);
        }
        __syncthreads();

        // ---- 8 K-steps of 8 WMMAs ----
#pragma unroll
        for (int kk = 0; kk < KC; kk += 4) {
            const int klo = kk + khalf;
            const v2f a0 = *(const v2f*)&A_s[(waveM +  0 + ml) * KPAD + klo];
            const v2f a1 = *(const v2f*)&A_s[(waveM + 16 + ml) * KPAD + klo];
            const v2f b0 = *(const v2f*)&B_s[(waveN +  0 + ml) * KPAD + klo];
            const v2f b1 = *(const v2f*)&B_s[(waveN + 16 + ml) * KPAD + klo];
            const v2f b2 = *(const v2f*)&B_s[(waveN + 32 + ml) * KPAD + klo];
            const v2f b3 = *(const v2f*)&B_s[(waveN + 48 + ml) * KPAD + klo];
            acc[0][0] = __builtin_amdgcn_wmma_f32_16x16x4_f32(false, a0, false, b0, (short)0, acc[0][0], false, false);
            acc[0][1] = __builtin_amdgcn_wmma_f32_16x16x4_f32(false, a0, false, b1, (short)0, acc[0][1], false, false);
            acc[0][2] = __builtin_amdgcn_wmma_f32_16x16x4_f32(false, a0, false, b2, (short)0, acc[0][2], false, false);
            acc[0][3] = __builtin_amdgcn_wmma_f32_16x16x4_f32(false, a0, false, b3, (short)0, acc[0][3], false, false);
            acc[1][0] = __builtin_amdgcn_wmma_f32_16x16x4_f32(false, a1, false, b0, (short)0, acc[1][0], false, false);
            acc[1][1] = __builtin_amdgcn_wmma_f32_16x16x4_f32(false, a1, false, b1, (short)0, acc[1][1], false, false);
            acc[1][2] = __builtin_amdgcn_wmma_f32_16x16x4_f32(false, a1, false, b2, (short)0, acc[1][2], false, false);
            acc[1][3] = __builtin_amdgcn_wmma_f32_16x16x4_f32(false, a1, false, b3, (short)0, acc[1][3], false, false);
        }
        __syncthreads();
    }

    // ---- epilogue: bias gather, log-freq correction, accidental-match mask ----
    const int mrow = (lane < 16) ? 0 : 8;
#pragma unroll
    for (int j = 0; j < 4; ++j) {
        const int c    = colBase + waveN + j * 16 + ml;
        const int sid  = sid_s[waveN + j * 16 + ml];
        const float cadj = bias[sid] - __logf(sample_freq[c]);
#pragma unroll
        for (int i = 0; i < 2; ++i) {
#pragma unroll
            for (int v = 0; v < 8; ++v) {
                const int rloc = waveM + i * 16 + v + mrow;
                float val = acc[i][j][v] + cadj;
                if (tgt_s[rloc] == sid) val = NEG_INF_V;
                logits[(size_t)(rowBase + rloc) * OUTSTRIDE + 1 + c] = val;
            }
        }
    }
}

// ---------------------------------------------------------------------------
// Column 0: true_logits[n] = dot(output[n], weight[targets[n]]) + bias[t]
//                            - log(true_freq[n])
// One wave32 per row; float4 loads; xor-shuffle reduction.
// ---------------------------------------------------------------------------
__global__ __launch_bounds__(256) void true_logits_kernel(
    const float* __restrict__ outp,
    const int*   __restrict__ targets,
    const float* __restrict__ true_freq,
    const float* __restrict__ weight,
    const float* __restrict__ bias,
    float*       __restrict__ logits)
{
    const int lane = threadIdx.x & 31;
    const int row  = blockIdx.x * 8 + (threadIdx.x >> 5);
    const int t    = targets[row];
    const float* a = outp   + (size_t)row * HIDDIM;
    const float* w = weight + (size_t)t   * HIDDIM;

    float s = 0.0f;
#pragma unroll
    for (int i = 0; i < HIDDIM / (32 * 4); ++i) {
        const v4f av = *(const v4f*)(a + (size_t)(i * 32 + lane) * 4);
        const v4f wv = *(const v4f*)(w + (size_t)(i * 32 + lane) * 4);
        s = fmaf(av.x, wv.x, s);
        s = fmaf(av.y, wv.y, s);
        s = fmaf(av.z, wv.z, s);
        s = fmaf(av.w, wv.w, s);
    }
#pragma unroll
    for (int off = 16; off > 0; off >>= 1) s += __shfl_xor(s, off, 32);
    if (lane == 0)
        logits[(size_t)row * OUTSTRIDE] = s + bias[t] - __logf(true_freq[row]);
}

// new_targets tail = zeros (int32 zero == float 0.0f bit pattern)
__global__ void zero_tail_kernel(float* __restrict__ p, int n)
{
    const int i = blockIdx.x * blockDim.x + threadIdx.x;
    if (i < n) p[i] = 0.0f;
}

extern "C" void kernel_launch(void* const* d_in, const int* in_sizes, int n_in,
                              void* d_out, int out_size, void* d_ws, size_t ws_size,
                              hipStream_t stream)
{
    const float* outp        = (const float*)d_in[0];
    const int*   targets     = (const int*)  d_in[1];
    const int*   sample_ids  = (const int*)  d_in[2];
    const float* true_freq   = (const float*)d_in[3];
    const float* sample_freq = (const float*)d_in[4];
    const float* weight      = (const float*)d_in[5];
    const float* bias        = (const float*)d_in[6];

    float* logits = (float*)d_out;
    float* tail   = logits + (size_t)NROWS * OUTSTRIDE;

    dim3 grid(NSAMP / BN, NROWS / BM);
    sampled_gemm_kernel<<<grid, 256, 0, stream>>>(outp, targets, sample_ids,
                                                  sample_freq, weight, bias, logits);
    true_logits_kernel<<<NROWS / 8, 256, 0, stream>>>(outp, targets, true_freq,
                                                      weight, bias, logits);
    zero_tail_kernel<<<NROWS / 256, 256, 0, stream>>>(tail, NROWS);
}